// SGC_49134425867022
// MI455X (gfx1250) — compile-verified
//
#include <hip/hip_runtime.h>

typedef float v2f __attribute__((ext_vector_type(2)));
typedef float v8f __attribute__((ext_vector_type(8)));

#define F_IN 128  // K dim of both GEMMs and feature width of both propagations

__device__ __forceinline__ void atomAddF(float* p, float v) {
    // guarantees global_atomic_add_f32 (no CAS loop)
    unsafeAtomicAdd(p, v);
}

// ---------------- degree / dinv ----------------

__global__ void sgc_init_deg(float* __restrict__ deg, int n) {
    int i = blockIdx.x * blockDim.x + threadIdx.x;
    if (i < n) deg[i] = 1.0f;  // self loop contributes 1 to every node's degree
}

__global__ void sgc_count_deg(float* __restrict__ deg, const int* __restrict__ dst, int nE) {
    int e = blockIdx.x * blockDim.x + threadIdx.x;
    if (e < nE) atomAddF(deg + dst[e], 1.0f);
}

__global__ void sgc_rsqrt(float* __restrict__ dinv, int n) {
    int i = blockIdx.x * blockDim.x + threadIdx.x;
    if (i < n) {
        float d = dinv[i];
        dinv[i] = d > 0.0f ? rsqrtf(d) : 0.0f;
    }
}

// ---------------- propagation ----------------

// OUT[i] = dinv[i]^2 * X[i]   (self-loop term; also zero-initializes OUT)
__global__ void sgc_selfloop(const float* __restrict__ X, float* __restrict__ OUT,
                             const float* __restrict__ dinv, int n) {
    int t = blockIdx.x * blockDim.x + threadIdx.x;   // n * 32 threads, float4 each
    int i = t >> 5;
    int lane = t & 31;
    if (i >= n) return;
    float nm = dinv[i];
    nm *= nm;
    float4 v = ((const float4*)(X + (size_t)i * F_IN))[lane];
    float4 o;
    o.x = v.x * nm; o.y = v.y * nm; o.z = v.z * nm; o.w = v.w * nm;
    ((float4*)(OUT + (size_t)i * F_IN))[lane] = o;
}

// one wave32 per edge: OUT[d] += dinv[s]*dinv[d] * X[s]
// edge id is wave-uniform -> force s, d, norm into SGPRs so the row loads use
// the scalar-base (SADDR) global_load form and the scale is scalar math.
__global__ void sgc_edge_prop(const float* __restrict__ X, float* __restrict__ OUT,
                              const float* __restrict__ dinv,
                              const int* __restrict__ src, const int* __restrict__ dst,
                              int nE) {
    int wave = (int)((blockIdx.x * (size_t)blockDim.x + threadIdx.x) >> 5);
    int lane = threadIdx.x & 31;
    if (wave >= nE) return;
    int s = __builtin_amdgcn_readfirstlane(src[wave]);
    int d = __builtin_amdgcn_readfirstlane(dst[wave]);
    float norm = dinv[s] * dinv[d];
    float4 v = ((const float4*)(X + (size_t)s * F_IN))[lane];   // global_load_b128, saddr
    float* od = OUT + (size_t)d * F_IN + lane * 4;
    atomAddF(od + 0, v.x * norm);
    atomAddF(od + 1, v.y * norm);
    atomAddF(od + 2, v.z * norm);
    atomAddF(od + 3, v.w * norm);
}

// ---------------- WMMA GEMM: OUT[n x FOUT] = X[n x 128] @ W[128 x FOUT] + bias ----------------
// One wave per 16x16 output tile, v_wmma_f32_16x16x4_f32 over K=128 (32 steps).
// FOUT is a compile-time constant so every B access is base + immediate offset.

template <int FOUT>
__global__ void sgc_gemm_wmma(const float* __restrict__ X, const float* __restrict__ W,
                              const float* __restrict__ bias, float* __restrict__ OUT) {
    constexpr int nTilesN = FOUT >> 4;
    int wave = (int)((blockIdx.x * (size_t)blockDim.x + threadIdx.x) >> 5);
    int lane = threadIdx.x & 31;

    int tm = (wave / nTilesN) << 4;
    int tn = (wave % nTilesN) << 4;

    int half = lane >> 4;                  // 0: K pair {0,1}; 1: K pair {2,3}
    int kk   = half << 1;
    int lidx = lane & 15;

    const float* xrow = X + (size_t)(tm + lidx) * F_IN + kk;   // A row (M = lidx)
    const float* wcol = W + (size_t)kk * FOUT + tn + lidx;     // B column (N = lidx)

    v8f c = {};
    #pragma unroll
    for (int k0 = 0; k0 < F_IN; k0 += 4) {
        v2f a = *(const v2f*)(xrow + k0);              // global_load_b64, imm offset
        v2f b;
        b.x = wcol[k0 * FOUT];                         // imm offsets (< 64 KB)
        b.y = wcol[k0 * FOUT + FOUT];
        c = __builtin_amdgcn_wmma_f32_16x16x4_f32(false, a, false, b,
                                                  (short)0, c, false, false);
    }

    float bv = bias[tn + lidx];
    int col = tn + lidx;
    int rowBase = tm + (half << 3);        // lanes 16-31 hold rows M=8..15
    #pragma unroll
    for (int v = 0; v < 8; ++v) {
        OUT[(size_t)(rowBase + v) * FOUT + col] = c[v] + bv;
    }
}

// ---------------- host ----------------

extern "C" void kernel_launch(void* const* d_in, const int* in_sizes, int n_in,
                              void* d_out, int out_size, void* d_ws, size_t ws_size,
                              hipStream_t stream) {
    const float* x   = (const float*)d_in[0];
    const int*   ei  = (const int*)d_in[1];
    const float* W1  = (const float*)d_in[2];
    const float* b1  = (const float*)d_in[3];
    const float* W2  = (const float*)d_in[4];
    const float* b2  = (const float*)d_in[5];
    float*       out = (float*)d_out;

    const int n  = in_sizes[0] / F_IN;     // 100000
    const int E  = in_sizes[1] / 2;        // 1600000

    const int* src = ei;                   // edge_index row-major (2, E)
    const int* dst = ei + E;

    // workspace: dinv | bufA (n*128) | bufB (n*128)
    size_t nAl = ((size_t)n + 127) & ~(size_t)127;
    float* dinv = (float*)d_ws;
    float* bufA = dinv + nAl;
    float* bufB = bufA + nAl * F_IN;

    const int B = 256;
    int gN  = (n + B - 1) / B;
    int gE  = (E + B - 1) / B;
    int gNF = (int)(((size_t)n * 32 + B - 1) / B);           // one float4 per lane
    int gEW = (int)(((size_t)E * 32 + B - 1) / B);           // one wave per edge

    // --- normalization ---
    sgc_init_deg<<<gN, B, 0, stream>>>(dinv, n);
    sgc_count_deg<<<gE, B, 0, stream>>>(dinv, dst, E);
    sgc_rsqrt<<<gN, B, 0, stream>>>(dinv, n);

    // --- layer 1: propagate x -> bufA, GEMM(128) -> bufB ---
    sgc_selfloop<<<gNF, B, 0, stream>>>(x, bufA, dinv, n);
    sgc_edge_prop<<<gEW, B, 0, stream>>>(x, bufA, dinv, src, dst, E);
    {
        int tiles = (n >> 4) * (128 >> 4);               // exact: 100000 % 16 == 0
        int grid  = (int)(((size_t)tiles * 32) / B);
        sgc_gemm_wmma<128><<<grid, B, 0, stream>>>(bufA, W1, b1, bufB);
    }

    // --- layer 2: propagate bufB -> bufA, GEMM(64) -> d_out ---
    sgc_selfloop<<<gNF, B, 0, stream>>>(bufB, bufA, dinv, n);
    sgc_edge_prop<<<gEW, B, 0, stream>>>(bufB, bufA, dinv, src, dst, E);
    {
        int tiles = (n >> 4) * (64 >> 4);
        int grid  = (int)(((size_t)tiles * 32) / B);
        sgc_gemm_wmma<64><<<grid, B, 0, stream>>>(bufA, W2, b2, out);
    }
}